// DepthwiseConv3D_80023830659695
// MI455X (gfx1250) — compile-verified
//
#include <hip/hip_runtime.h>
#include <hip/hip_bf16.h>

// Depthwise 3x3 conv, NHWDC fp32, SAME padding, + bias.
// x: (8,64,64,32,64), w: (32,3,3,64), b: (32,64)  -> y: (8,64,64,32,64)
// Bandwidth-bound (AI ~2.25 FLOP/B); staged through LDS via CDNA5 async copies.

#define BB 8
#define HH 64
#define WW 64
#define DD 32
#define CC 64

// strides in floats
#define S_B (HH * WW * DD * CC) // 8388608
#define S_H (WW * DD * CC)      // 131072
#define S_W (DD * CC)           // 2048
#define S_D (CC)                // 64

typedef float v4f __attribute__((ext_vector_type(4)));

__global__ __launch_bounds__(256) void dwconv3x3_kernel(
    const float* __restrict__ x,
    const float* __restrict__ w,
    const float* __restrict__ bias,
    float* __restrict__ y)
{
    // 3 input rows of (W=64, C=64) fp32 = 48 KB, + weights (9*64) + bias (64)
    __shared__ v4f lds_x[3 * 1024]; // [row][w*16 + c4]
    __shared__ v4f lds_w[144];      // [(kh*3+kw)*16 + c4]
    __shared__ v4f lds_b[16];       // [c4]

    const int t   = threadIdx.x;      // 0..255
    const int bid = blockIdx.x;       // B*D*H blocks, h fastest for L2 row reuse
    const int h   = bid % HH;
    const int d   = (bid / HH) % DD;
    const int b   = bid / (HH * DD);
    const int c4  = t & 15;           // channel quad, fixed per thread
    const int wt  = t >> 4;           // 0..15, w-position base

    // ---- stage 3 input rows into LDS with async global->LDS copies ----
    #pragma unroll
    for (int r = 0; r < 3; ++r) {
        const int hh = h - 1 + r;
        if (hh >= 0 && hh < HH) {
            const float* rowbase = x + (size_t)b * S_B + (size_t)hh * S_H
                                     + (size_t)d * S_D + (size_t)c4 * 4;
            #pragma unroll
            for (int k = 0; k < 4; ++k) {
                const int wcol = wt + 16 * k;
                unsigned long long gaddr =
                    (unsigned long long)(uintptr_t)(rowbase + (size_t)wcol * S_W);
                unsigned lds_addr =
                    (unsigned)(uintptr_t)(&lds_x[r * 1024 + t + 256 * k]);
                // CDNA5 async tensor/LDS path; tracked by ASYNCcnt.
                asm volatile("global_load_async_to_lds_b128 %0, %1, off"
                             :: "v"(lds_addr), "v"(gaddr)
                             : "memory");
            }
        } else {
            const v4f z = (v4f)0.0f;
            #pragma unroll
            for (int k = 0; k < 4; ++k)
                lds_x[r * 1024 + t + 256 * k] = z;
        }
    }

    // ---- weights + bias for this d via regular loads (tiny: 2.5 KB) ----
    if (t < 144) {
        lds_w[t] = ((const v4f*)(w + (size_t)d * (9 * CC)))[t];
    } else if (t < 160) {
        lds_b[t - 144] = ((const v4f*)(bias + (size_t)d * CC))[t - 144];
    }

    // L2 prefetch of the fresh row the next block (h+1) will need (hh = h+2).
    {
        const int hp = h + 2;
        if (hp < HH) {
            const float* p = x + (size_t)b * S_B + (size_t)hp * S_H
                               + (size_t)d * S_D + (size_t)c4 * 4
                               + (size_t)wt * S_W;
            __builtin_prefetch(p, 0, 1); // -> global_prefetch_b8
        }
    }

    // Wait for this wave's async copies, then workgroup-sync (LDS visible).
    asm volatile("s_wait_asynccnt 0" ::: "memory");
    __syncthreads();

    // ---- hoist the 9 weight quads + bias quad into registers ----
    v4f wv[9];
    #pragma unroll
    for (int j = 0; j < 9; ++j)
        wv[j] = lds_w[j * 16 + c4];
    const v4f bv = lds_b[c4];

    // ---- compute 4 output quads per thread ----
    #pragma unroll
    for (int k = 0; k < 4; ++k) {
        const int w0 = wt + 16 * k; // 0..63
        v4f acc = bv;
        #pragma unroll
        for (int kh = 0; kh < 3; ++kh) {
            #pragma unroll
            for (int kw = 0; kw < 3; ++kw) {
                const int ww2 = w0 + kw - 1;
                if (ww2 >= 0 && ww2 < WW) {
                    const v4f xv = lds_x[kh * 1024 + ww2 * 16 + c4];
                    acc += xv * wv[kh * 3 + kw];
                }
            }
        }
        v4f* outp = (v4f*)(y + (size_t)b * S_B + (size_t)h * S_H
                             + (size_t)w0 * S_W + (size_t)d * S_D
                             + (size_t)c4 * 4);
        __builtin_nontemporal_store(acc, outp); // output never re-read
    }
}

extern "C" void kernel_launch(void* const* d_in, const int* in_sizes, int n_in,
                              void* d_out, int out_size, void* d_ws, size_t ws_size,
                              hipStream_t stream)
{
    (void)in_sizes; (void)n_in; (void)out_size; (void)d_ws; (void)ws_size;
    const float* x    = (const float*)d_in[0];
    const float* w    = (const float*)d_in[1];
    const float* bias = (const float*)d_in[2];
    float*       y    = (float*)d_out;

    const int grid = BB * DD * HH; // 16384 blocks, one output row (b,d,h) each
    dwconv3x3_kernel<<<grid, 256, 0, stream>>>(x, w, bias, y);
}